// MPS_RNN_2D_64269890617470
// MI455X (gfx1250) — compile-verified
//
#include <hip/hip_runtime.h>
#include <math.h>

namespace {

constexpr int MDIM = 8, LDIM = 8, DC = 6, HLN = 2, NQ = 64;

typedef float v2f __attribute__((ext_vector_type(2)));
typedef float v8f __attribute__((ext_vector_type(8)));

// Swap data between lane n and lane n+16 (SWAPX16: xor_mask=0x10, and=0x1f)
__device__ __forceinline__ float swapx16(float x) {
  return __int_as_float(__builtin_amdgcn_ds_swizzle(__float_as_int(x), 0x401F));
}

// Native-rate helpers (v_rcp_f32 / v_sqrt_f32 / v_rsq_f32), no IEEE fixup chains
__device__ __forceinline__ float nrcp(float x)  { return __builtin_amdgcn_rcpf(x); }
__device__ __forceinline__ float nsqrt(float x) { return __builtin_amdgcn_sqrtf(x); }
__device__ __forceinline__ float nrsq(float x)  { return __builtin_amdgcn_rsqf(x); }

// atan2 via one native rcp + deg-11 minimax atan on [0,1] (~2e-6 max err)
__device__ __forceinline__ float fast_atan2(float y, float x) {
  const float PI   = 3.14159265358979f;
  const float PIO2 = 1.57079632679490f;
  float ax = fabsf(x), ay = fabsf(y);
  float mx = fmaxf(ax, ay), mn = fminf(ax, ay);
  float a  = mn * nrcp(mx);
  a = (mx == 0.0f) ? 0.0f : a;           // atan2(0,0) -> 0
  float s = a * a;
  float p = -0.0117212f;
  p = fmaf(p, s,  0.05265332f);
  p = fmaf(p, s, -0.11643287f);
  p = fmaf(p, s,  0.19354346f);
  p = fmaf(p, s, -0.33262347f);
  p = fmaf(p, s,  0.99997726f);
  float r = p * a;
  r = (ay > ax)   ? (PIO2 - r) : r;      // reflect across pi/4
  r = (x < 0.0f)  ? (PI - r)   : r;      // left half-plane
  r = (y < 0.0f)  ? -r         : r;      // lower half-plane
  return r;
}

enum HHMode { HH_ONES, HH_ZEROS, HH_AM1, HH_AP1 };

struct Ctx {
  const float *pMh, *pMv, *pv, *pwr, *pcr, *peta;
  unsigned long long tbits;  // target spins of this lane's sample, bit i = site i
  int H;      // lane>>4 : which HL component this lane owns (C/D/B row half)
  int hm;     // (lane&15)>>3 : A-matrix row h index
  int cm;     // lane&7      : A-matrix row c index
  int kb;     // H*2         : K offset inside a K=4 WMMA step
  bool cmOK;  // cm < DC
};

template <int A, HHMode MODE>
__device__ __forceinline__ void site(const Ctx& c, int b, int i,
                                     float (&row)[MDIM][DC],
                                     float& amp, float& phi) {
  const int k = (i > 0) ? (i - 1) : 0;
  const bool takeOwn = ((int)((c.tbits >> k) & 1ull)) == c.H;

  // hv: vertical neighbor h[a][b-1] (zero on first row), HL-gathered by spin q
  float hv[DC];
#pragma unroll
  for (int d = 0; d < DC; ++d) {
    float own = row[A][d];
    float par = swapx16(own);
    float sel = takeOwn ? own : par;
    hv[d] = (b == 0) ? 0.0f : sel;
  }

  // hh: horizontal neighbor per reference's exact rules (compile-time mode)
  float hh[DC];
  if constexpr (MODE == HH_ONES) {
#pragma unroll
    for (int d = 0; d < DC; ++d) hh[d] = 1.0f;
  } else if constexpr (MODE == HH_ZEROS) {
#pragma unroll
    for (int d = 0; d < DC; ++d) hh[d] = 0.0f;
  } else {
    constexpr int AN = (MODE == HH_AM1) ? (A - 1) : (A + 1);
#pragma unroll
    for (int d = 0; d < DC; ++d) {
      float own = row[AN][d];
      float par = swapx16(own);
      hh[d] = takeOwn ? own : par;
    }
  }

  const int so = A * LDIM + b;

  // B operands (4x16, K x N): VGPR0 = K{0|2}, VGPR1 = K{1|3} by lane half.
  const bool lo = (c.H == 0);
  v2f Bv0, Bv1, Bh0, Bh1;
  Bv0.x = lo ? hv[0] : hv[2];  Bv0.y = lo ? hv[1] : hv[3];
  Bv1.x = lo ? hv[4] : 0.0f;   Bv1.y = lo ? hv[5] : 0.0f;   // K=6,7 pad
  Bh0.x = lo ? hh[0] : hh[2];  Bh0.y = lo ? hh[1] : hh[3];
  Bh1.x = lo ? hh[4] : 0.0f;   Bh1.y = lo ? hh[5] : 0.0f;

  // A operands (16x4): A[row=(h*8+cm), K] = M[h][K][cm]; rows cm>=6 and K>=6 zero
  const float* bV = c.pMv + so * (HLN * DC * DC);
  const float* bH = c.pMh + so * (HLN * DC * DC);
  auto ldA = [&](const float* base, int kk) -> float {
    return (c.cmOK && kk < DC) ? base[c.hm * (DC * DC) + kk * DC + c.cm] : 0.0f;
  };
  v2f Av0, Av1, Ah0, Ah1;
  Av0.x = ldA(bV, c.kb);     Av0.y = ldA(bV, c.kb + 1);
  Av1.x = ldA(bV, c.kb + 4); Av1.y = ldA(bV, c.kb + 5);
  Ah0.x = ldA(bH, c.kb);     Ah0.y = ldA(bH, c.kb + 1);
  Ah1.x = ldA(bH, c.kb + 4); Ah1.y = ldA(bH, c.kb + 5);

  // Fold residual + bias + initial-ones into the accumulator:
  // h_ud = 1 + hv + hh + v  +  Mv@hv  +  Mh@hh
  const float* vb = c.pv + so * (HLN * DC) + c.H * DC;
  v8f acc;
#pragma unroll
  for (int d = 0; d < DC; ++d) acc[d] = 1.0f + hv[d] + hh[d] + vb[d];
  acc[6] = 0.0f;
  acc[7] = 0.0f;

  acc = __builtin_amdgcn_wmma_f32_16x16x4_f32(false, Av0, false, Bv0, (short)0, acc, false, false);
  acc = __builtin_amdgcn_wmma_f32_16x16x4_f32(false, Av1, false, Bv1, (short)0, acc, false, false);
  acc = __builtin_amdgcn_wmma_f32_16x16x4_f32(false, Ah0, false, Bh0, (short)0, acc, false, false);
  acc = __builtin_amdgcn_wmma_f32_16x16x4_f32(false, Ah1, false, Bh1, (short)0, acc, false, false);

  // Normalize over both HL components (lane-pair reduction)
  float s2 = 0.0f;
#pragma unroll
  for (int d = 0; d < DC; ++d) s2 = fmaf(acc[d], acc[d], s2);
  float tot = s2 + swapx16(s2);
  float inv = nrsq(tot);

  float hud[DC];
#pragma unroll
  for (int d = 0; d < DC; ++d) {
    hud[d] = acc[d] * inv;
    row[A][d] = hud[d];
  }

  // P[H] = sqrt(ah^T |eta| ah)  (per-component quadratic form, eta wave-uniform)
  const float* eb = c.peta + so * (DC * DC);
  float ah[DC];
#pragma unroll
  for (int d = 0; d < DC; ++d) ah[d] = fabsf(hud[d]);
  float quad = 0.0f;
#pragma unroll
  for (int bb = 0; bb < DC; ++bb) {
    float u = 0.0f;
#pragma unroll
    for (int aa = 0; aa < DC; ++aa) u = fmaf(ah[aa], fabsf(eb[aa * DC + bb]), u);
    quad = fmaf(u, ah[bb], quad);
  }
  float P = nsqrt(quad);
  float Pp = swapx16(P);
  float rmx = nrcp(fmaxf(P, Pp));
  float Pn = P * rmx;
  float Ppn = Pp * rmx;
  float rnn = nrcp(fmaxf(nsqrt(fmaf(Pn, Pn, Ppn * Ppn)), 1e-15f));
  bool tOwn = ((int)((c.tbits >> i) & 1ull)) == c.H;
  amp *= (tOwn ? Pn : Ppn) * rnn;

  // Phase: h_i = HL-gather(h_ud, q); re/im = w_r . h_i + c_r
  const float* wb = c.pwr + so * (DC * 2);
  float re = c.pcr[so * 2 + 0];
  float im = c.pcr[so * 2 + 1];
#pragma unroll
  for (int d = 0; d < DC; ++d) {
    float own = hud[d];
    float par = swapx16(own);
    float hi = takeOwn ? own : par;
    re = fmaf(wb[d * 2 + 0], hi, re);
    im = fmaf(wb[d * 2 + 1], hi, im);
  }
  phi += fast_atan2(im, re);
}

}  // namespace

__global__ void __launch_bounds__(256)
mps_rnn2d_wmma(const int* __restrict__ x,
               const float* __restrict__ pMh, const float* __restrict__ pMv,
               const float* __restrict__ pv, const float* __restrict__ pwr,
               const float* __restrict__ pcr, const float* __restrict__ peta,
               float* __restrict__ out, int nb) {
  const int tid = blockIdx.x * blockDim.x + threadIdx.x;
  const int lane = threadIdx.x & 31;
  const int wid = tid >> 5;
  // Wave-uniform guard: whole wave exits together so EXEC is all-ones at WMMAs
  if (wid * 16 >= nb) return;
  const int n16 = lane & 15;
  const int sample = wid * 16 + n16;

  Ctx c;
  c.pMh = pMh; c.pMv = pMv; c.pv = pv; c.pwr = pwr; c.pcr = pcr; c.peta = peta;
  c.H = lane >> 4;
  c.hm = (lane & 15) >> 3;
  c.cm = lane & 7;
  c.kb = c.H * 2;
  c.cmOK = (c.cm < DC);

  // Pack this sample's 64 spins into a bitmask (lane n and n+16 load the same row)
  const int* xr = x + sample * NQ;
  unsigned long long tb = 0ull;
#pragma unroll
  for (int j = 0; j < NQ; ++j)
    tb |= (unsigned long long)(xr[j] > 0 ? 1u : 0u) << j;
  c.tbits = tb;

  float row[MDIM][DC];
#pragma unroll
  for (int a = 0; a < MDIM; ++a)
#pragma unroll
    for (int d = 0; d < DC; ++d) row[a][d] = 1.0f;

  float amp = 1.0f, phi = 0.0f;

#pragma unroll 1
  for (int b = 0; b < LDIM; ++b) {
    const int i0 = b * MDIM;
    if ((b & 1) == 0) {  // snake left->right
      site<0, HH_ONES >(c, b, i0 + 0, row, amp, phi);
      site<1, HH_ONES >(c, b, i0 + 1, row, amp, phi);
      site<2, HH_AM1  >(c, b, i0 + 2, row, amp, phi);
      site<3, HH_ONES >(c, b, i0 + 3, row, amp, phi);
      site<4, HH_AM1  >(c, b, i0 + 4, row, amp, phi);
      site<5, HH_ONES >(c, b, i0 + 5, row, amp, phi);
      site<6, HH_AM1  >(c, b, i0 + 6, row, amp, phi);
      site<7, HH_ZEROS>(c, b, i0 + 7, row, amp, phi);
    } else {             // snake right->left
      site<7, HH_ZEROS>(c, b, i0 + 0, row, amp, phi);
      site<6, HH_ONES >(c, b, i0 + 1, row, amp, phi);
      site<5, HH_AP1  >(c, b, i0 + 2, row, amp, phi);
      site<4, HH_ONES >(c, b, i0 + 3, row, amp, phi);
      site<3, HH_AP1  >(c, b, i0 + 4, row, amp, phi);
      site<2, HH_ONES >(c, b, i0 + 5, row, amp, phi);
      site<1, HH_AP1  >(c, b, i0 + 6, row, amp, phi);
      site<0, HH_ONES >(c, b, i0 + 7, row, amp, phi);
    }
  }

  // complex64 output: interleaved (re, im); one writer per sample
  if (c.H == 0) {
    out[2 * sample + 0] = amp * __cosf(phi);
    out[2 * sample + 1] = amp * __sinf(phi);
  }
}

extern "C" void kernel_launch(void* const* d_in, const int* in_sizes, int n_in,
                              void* d_out, int out_size, void* d_ws, size_t ws_size,
                              hipStream_t stream) {
  const int*   x    = (const int*)d_in[0];
  const float* pMh  = (const float*)d_in[1];
  const float* pMv  = (const float*)d_in[2];
  const float* pv   = (const float*)d_in[3];
  const float* pwr  = (const float*)d_in[4];
  const float* pcr  = (const float*)d_in[5];
  const float* peta = (const float*)d_in[6];

  const int nb = in_sizes[0] / NQ;             // 16384
  const int waves = (nb + 15) / 16;            // 16 samples per wave32
  const int threads = waves * 32;
  const int block = 256;
  const int grid = (threads + block - 1) / block;

  mps_rnn2d_wmma<<<grid, block, 0, stream>>>(x, pMh, pMv, pv, pwr, pcr, peta,
                                             (float*)d_out, nb);
}